// ProteinMapper_44375602102606
// MI455X (gfx1250) — compile-verified
//
#include <hip/hip_runtime.h>
#include <hip/hip_bf16.h>

// ---------------------------------------------------------------------------
// Shapes from the reference
#define NRES 21
#define NGRP 8
#define NATM 14

typedef unsigned int u32x4 __attribute__((ext_vector_type(4)));
typedef int          i32x8 __attribute__((ext_vector_type(8)));
typedef int          i32x4 __attribute__((ext_vector_type(4)));

// ---------------------------------------------------------------------------
// TDM: DMA `n` dwords from global memory into LDS (1-D tile descriptor).
// Tracked with TENSORcnt. Descriptor layout per CDNA5 ISA sec. 8 (D# groups).
// This toolchain (clang-23 / therock-10.0) exposes the 6-arg builtin:
//   (u32x4 g0, i32x8 g1, i32x4 g2, i32x4 g3, i32x8 extra, i32 cpol)
__device__ __forceinline__ void tdm_load_dwords_to_lds(const void* gptr,
                                                       const void* lptr,
                                                       unsigned n) {
  unsigned long long ga = (unsigned long long)gptr;
  // generic pointer to LDS: low 32 bits are the LDS byte offset (aperture in
  // the high word) -- see ISA 10.2 aperture address calculation.
  unsigned lds_addr = (unsigned)(unsigned long long)lptr;

  u32x4 g0;
  g0.x = 1u;                                       // count=1 (valid), user desc
  g0.y = lds_addr;                                 // lds_addr (bytes)
  g0.z = (unsigned)(ga & 0xFFFFFFFFull);           // global_addr[31:0]
  g0.w = (unsigned)((ga >> 32) & 0x1FFFFFFull)     // global_addr[56:32]
       | 0x80000000u;                              // type = 2 ("image")

  i32x8 g1;
  g1[0] = (int)(2u << 16);                         // data_size = 4 bytes; wg_mask=0
  g1[1] = (int)((n & 0xFFFFu) << 16);              // tensor_dim0[15:0]
  g1[2] = (int)(((n >> 16) & 0xFFFFu)              // tensor_dim0[31:16]
              | (1u << 16));                       // tensor_dim1[15:0] = 1
  g1[3] = (int)((n & 0xFFFFu) << 16);              // tensor_dim1[31:16]=0, tile_dim0 = n
  g1[4] = 1;                                       // tile_dim1 = 1, tile_dim2 = 0
  g1[5] = (int)n;                                  // tensor_dim0_stride[31:0] = n
  g1[6] = 0;                                       // stride0 hi, stride1 lo
  g1[7] = 0;                                       // stride1 hi

  i32x4 z4 = {0, 0, 0, 0};                         // groups 2/3 unused (<=2D)
  i32x8 z8 = {0, 0, 0, 0, 0, 0, 0, 0};             // extra arg (6-arg form)
  __builtin_amdgcn_tensor_load_to_lds(g0, g1, z4, z4, z8, 0);
}

// ---------------------------------------------------------------------------
// One thread per (residue, atom). Lookup tables staged in LDS via TDM.
__global__ __launch_bounds__(256)
void protein_atoms_kernel(const float* __restrict__ bb_rots,   // (R,3,3)
                          const float* __restrict__ bb_trans,  // (R,3)
                          const float* __restrict__ angles,    // (R,7,2)
                          const int*   __restrict__ aatype,    // (R,)
                          const float* __restrict__ dframes,   // (21,8,4,4)
                          const int*   __restrict__ gidx,      // (21,14)
                          const float* __restrict__ amask,     // (21,14)
                          const float* __restrict__ litpos,    // (21,14,3)
                          float*       __restrict__ out,       // (R,14,3)
                          int nthreads)                        // R*14
{
  __shared__ float s_fr[NRES * NGRP * 16];   // 2688 dwords: default 4x4 frames
  __shared__ int   s_gi[NRES * NATM];        //  294 dwords: group indices
  __shared__ float s_am[NRES * NATM];        //  294 dwords: atom mask
  __shared__ float s_lp[NRES * NATM * 3];    //  882 dwords: literature positions

  // Wave 0 issues the table DMAs (wave-uniform branch; TDM ignores EXEC).
  if ((threadIdx.x >> 5) == 0) {
    tdm_load_dwords_to_lds(dframes, s_fr, NRES * NGRP * 16);
    tdm_load_dwords_to_lds(gidx,    s_gi, NRES * NATM);
    tdm_load_dwords_to_lds(amask,   s_am, NRES * NATM);
    tdm_load_dwords_to_lds(litpos,  s_lp, NRES * NATM * 3);
    __builtin_amdgcn_s_wait_tensorcnt(0);
  }
  __syncthreads();

  int t = blockIdx.x * 256 + threadIdx.x;
  if (t >= nthreads) return;

  unsigned r = (unsigned)t / (unsigned)NATM;     // residue
  int      a = t - (int)(r * NATM);              // atom within residue

  int aa = aatype[r];

  // Backbone frame: M = bb_rots[r], v = bb_trans[r]
  const float* bp = bb_rots + (size_t)r * 9;
  float M0 = bp[0], M1 = bp[1], M2 = bp[2];
  float M3 = bp[3], M4 = bp[4], M5 = bp[5];
  float M6 = bp[6], M7 = bp[7], M8 = bp[8];
  const float* tp = bb_trans + (size_t)r * 3;
  float v0 = tp[0], v1 = tp[1], v2 = tp[2];

  int g = s_gi[aa * NATM + a];                   // rigid group of this atom

  const float* fb = s_fr + aa * (NGRP * 16);
  const float* ap = angles + (size_t)r * 14;

  // Kinematic chain in the global frame:
  //   g <= 4 : global = bb o f_g                (single application)
  //   g >= 5 : global = bb o f4 o f5 o ... o f_g (groups 5..7 chain off 4)
  int k0 = (g <= 4) ? g : 4;
  for (int k = k0; k <= g; ++k) {
    float s, c;
    if (k == 0) { s = 0.0f; c = 1.0f; }
    else        { s = ap[(k - 1) * 2]; c = ap[(k - 1) * 2 + 1]; }

    const float* f = fb + k * 16;                // row-major 4x4
    float d0 = f[0], d1 = f[1],  d2 = f[2],  e0 = f[3];
    float d3 = f[4], d4 = f[5],  d5 = f[6],  e1 = f[7];
    float d6 = f[8], d7 = f[9],  d8 = f[10], e2 = f[11];

    // F = dR @ rotx,  rotx = [[1,0,0],[0,c,-s],[0,s,c]]
    float F0 = d0,                 F1 = fmaf(d1, c, d2 * s), F2 = fmaf(d2, c, -d1 * s);
    float F3 = d3,                 F4 = fmaf(d4, c, d5 * s), F5 = fmaf(d5, c, -d4 * s);
    float F6 = d6,                 F7 = fmaf(d7, c, d8 * s), F8 = fmaf(d8, c, -d7 * s);

    // v' = M @ e + v   (use old M)
    float nv0 = fmaf(M0, e0, fmaf(M1, e1, fmaf(M2, e2, v0)));
    float nv1 = fmaf(M3, e0, fmaf(M4, e1, fmaf(M5, e2, v1)));
    float nv2 = fmaf(M6, e0, fmaf(M7, e1, fmaf(M8, e2, v2)));

    // M' = M @ F
    float N0 = fmaf(M0, F0, fmaf(M1, F3, M2 * F6));
    float N1 = fmaf(M0, F1, fmaf(M1, F4, M2 * F7));
    float N2 = fmaf(M0, F2, fmaf(M1, F5, M2 * F8));
    float N3 = fmaf(M3, F0, fmaf(M4, F3, M5 * F6));
    float N4 = fmaf(M3, F1, fmaf(M4, F4, M5 * F7));
    float N5 = fmaf(M3, F2, fmaf(M4, F5, M5 * F8));
    float N6 = fmaf(M6, F0, fmaf(M7, F3, M8 * F6));
    float N7 = fmaf(M6, F1, fmaf(M7, F4, M8 * F7));
    float N8 = fmaf(M6, F2, fmaf(M7, F5, M8 * F8));

    M0 = N0; M1 = N1; M2 = N2;
    M3 = N3; M4 = N4; M5 = N5;
    M6 = N6; M7 = N7; M8 = N8;
    v0 = nv0; v1 = nv1; v2 = nv2;
  }

  // Atom placement: pred = (M @ lit + v) * mask
  int   ai = aa * NATM + a;
  float lx = s_lp[ai * 3 + 0], ly = s_lp[ai * 3 + 1], lz = s_lp[ai * 3 + 2];
  float m  = s_am[ai];

  float px = fmaf(M0, lx, fmaf(M1, ly, fmaf(M2, lz, v0))) * m;
  float py = fmaf(M3, lx, fmaf(M4, ly, fmaf(M5, lz, v1))) * m;
  float pz = fmaf(M6, lx, fmaf(M7, ly, fmaf(M8, lz, v2))) * m;

  float* o = out + (size_t)t * 3;
  o[0] = px; o[1] = py; o[2] = pz;
}

// ---------------------------------------------------------------------------
extern "C" void kernel_launch(void* const* d_in, const int* in_sizes, int n_in,
                              void* d_out, int out_size, void* d_ws, size_t ws_size,
                              hipStream_t stream) {
  const float* bb_rots  = (const float*)d_in[0];
  const float* bb_trans = (const float*)d_in[1];
  const float* angles   = (const float*)d_in[2];
  const int*   aatype   = (const int*)d_in[3];
  const float* dframes  = (const float*)d_in[4];
  const int*   gidx     = (const int*)d_in[5];
  const float* amask    = (const float*)d_in[6];
  const float* litpos   = (const float*)d_in[7];
  float*       out      = (float*)d_out;

  int nres     = in_sizes[3];        // B*L residues
  int nthreads = nres * NATM;        // one thread per (residue, atom)
  int nblocks  = (nthreads + 255) / 256;

  protein_atoms_kernel<<<nblocks, 256, 0, stream>>>(
      bb_rots, bb_trans, angles, aatype, dframes, gidx, amask, litpos,
      out, nthreads);
}